// SeparationLoss_51513837748796
// MI455X (gfx1250) — compile-verified
//
#include <hip/hip_runtime.h>
#include <hip/hip_bf16.h>

typedef __attribute__((ext_vector_type(2))) float v2f;
typedef __attribute__((ext_vector_type(8))) float v8f;

#define B_ 8
#define L_ 64
#define K_ 32
#define C_ 32
#define D_ 1024
#define PAIRS (B_ * L_ * L_)   // 32768

// ---------- wave32 reduction helpers ----------
__device__ inline float wave_sum32(float v) {
  #pragma unroll
  for (int off = 16; off > 0; off >>= 1) v += __shfl_xor(v, off, 32);
  return v;
}
__device__ inline float wave_max32(float v) {
  #pragma unroll
  for (int off = 16; off > 0; off >>= 1) v = fmaxf(v, __shfl_xor(v, off, 32));
  return v;
}

// ---------- K1: log_softmax over C=32, one wave per (b,l,k) row ----------
__global__ __launch_bounds__(256) void k_softmax(const float* __restrict__ prior,
                                                 float* __restrict__ P,
                                                 float* __restrict__ LP) {
  int wid  = (blockIdx.x * blockDim.x + threadIdx.x) >> 5;  // row id, 0..16383
  int lane = threadIdx.x & 31;                               // c index
  if (wid >= B_ * L_ * K_) return;
  int idx = wid * C_ + lane;
  float x = prior[idx];
  float mx = wave_max32(x);
  float e = __expf(x - mx);
  float s = wave_sum32(e);
  float lp = x - mx - __logf(s);
  LP[idx] = lp;
  P[idx]  = __expf(lp);
}

// ---------- K2: squared norms of h rows, one wave per (b,l) ----------
__global__ __launch_bounds__(256) void k_sqnorm(const float* __restrict__ h,
                                                float* __restrict__ SQ) {
  int wid  = (blockIdx.x * blockDim.x + threadIdx.x) >> 5;  // 0..511
  int lane = threadIdx.x & 31;
  if (wid >= B_ * L_) return;
  const float* row = h + wid * D_;
  float acc = 0.f;
  #pragma unroll 4
  for (int t = lane; t < D_; t += 32) {
    float v = row[t];
    acc = fmaf(v, v, acc);
  }
  acc = wave_sum32(acc);
  if (lane == 0) SQ[wid] = acc;
}

// ---------- K3: batched Gram G = H * H^T via V_WMMA_F32_16X16X4_F32 ----------
// One wave per 16x16 output tile: 8 batches * 4*4 tiles = 128 blocks of 32.
__global__ __launch_bounds__(32) void k_gram_wmma(const float* __restrict__ h,
                                                  float* __restrict__ G) {
  int blk = blockIdx.x;        // 0..127
  int b   = blk >> 4;
  int ti  = (blk >> 2) & 3;
  int tj  = blk & 3;
  int lane = threadIdx.x;      // 0..31

  // 32-bit A 16x4 layout: lanes 0-15 -> M=0..15 with K={0,1}; lanes 16-31 -> K={2,3}.
  // B 4x16 mirrors this with lanes as N. Both gathers read H[tile*16 + (lane&15), kk+khalf..+1].
  int r     = lane & 15;
  int khalf = (lane >> 4) << 1;   // 0 or 2
  const float* hb   = h + b * L_ * D_;
  const float* arow = hb + (ti * 16 + r) * D_;
  const float* brow = hb + (tj * 16 + r) * D_;

  v8f c = {0.f, 0.f, 0.f, 0.f, 0.f, 0.f, 0.f, 0.f};
  for (int kk = 0; kk < D_; kk += 4) {
    v2f a, bb;
    a.x  = arow[kk + khalf];
    a.y  = arow[kk + khalf + 1];
    bb.x = brow[kk + khalf];
    bb.y = brow[kk + khalf + 1];
    c = __builtin_amdgcn_wmma_f32_16x16x4_f32(
        /*neg_a=*/false, a, /*neg_b=*/false, bb,
        /*c_mod=*/(short)0, c, /*reuse_a=*/false, /*reuse_b=*/false);
  }

  // 16x16 f32 C/D layout: VGPR v -> M=v (lanes 0-15), M=v+8 (lanes 16-31); N = lane&15.
  float* gout = G + b * L_ * L_;
  int mhalf = (lane >> 4) << 3;   // 0 or 8
  int col   = tj * 16 + (lane & 15);
  #pragma unroll
  for (int v = 0; v < 8; ++v) {
    int row = ti * 16 + v + mhalf;
    gout[row * L_ + col] = c[v];
  }
}

// ---------- K4: per-pair JSD + masked att/rep terms. One wave per (b,i,j) ----------
__global__ __launch_bounds__(256) void k_pairs(const float* __restrict__ P,
                                               const float* __restrict__ LP,
                                               const float* __restrict__ SQ,
                                               const float* __restrict__ G,
                                               float* __restrict__ ATT,
                                               float* __restrict__ REP,
                                               float* __restrict__ AM,
                                               float* __restrict__ RM) {
  int wid  = (blockIdx.x * blockDim.x + threadIdx.x) >> 5;  // pair id, 0..32767
  int lane = threadIdx.x & 31;                               // c index
  if (wid >= PAIRS) return;
  int b = wid >> 12;
  int i = (wid >> 6) & 63;
  int j = wid & 63;

  if (j <= i) {   // strictly-upper-triangular only (uniform across the wave)
    if (lane == 0) { ATT[wid] = 0.f; REP[wid] = 0.f; AM[wid] = 0.f; RM[wid] = 0.f; }
    return;
  }

  const float* Pi  = P  + ((b * L_ + i) * K_) * C_;
  const float* Pj  = P  + ((b * L_ + j) * K_) * C_;
  const float* LPi = LP + ((b * L_ + i) * K_) * C_;
  const float* LPj = LP + ((b * L_ + j) * K_) * C_;

  float acc = 0.f;
  #pragma unroll 4
  for (int k = 0; k < K_; ++k) {
    int o = k * C_ + lane;
    float p1  = Pi[o],  p2  = Pj[o];
    float lp1 = LPi[o], lp2 = LPj[o];
    float logm = __logf(0.5f * (p1 + p2));
    acc += p1 * (lp1 - logm) + p2 * (lp2 - logm);
  }
  acc = wave_sum32(acc);
  float jsd = 0.5f * acc * (1.0f / (float)K_);   // * SCALING_FACTOR (=1)

  if (lane == 0) {
    float mse = (SQ[b * L_ + i] + SQ[b * L_ + j]
                 - 2.0f * G[(b * L_ + i) * L_ + j]) * (1.0f / (float)D_);
    bool att = (jsd < 0.5f);
    ATT[wid] = att ? mse : 0.f;
    REP[wid] = att ? 0.f : __expf(-mse);         // EXP_TMP = 1
    AM[wid]  = att ? 1.f : 0.f;
    RM[wid]  = att ? 0.f : 1.f;
  }
}

// ---------- K5: deterministic fixed-order reduction to the two outputs ----------
__global__ __launch_bounds__(1024) void k_reduce(const float* __restrict__ ATT,
                                                 const float* __restrict__ REP,
                                                 const float* __restrict__ AM,
                                                 const float* __restrict__ RM,
                                                 float* __restrict__ out) {
  __shared__ float sA[1024], sR[1024], sM[1024], sN[1024];
  int t = threadIdx.x;
  float aA = 0.f, aR = 0.f, aM = 0.f, aN = 0.f;
  for (int idx = t; idx < PAIRS; idx += 1024) {   // fixed per-thread order
    aA += ATT[idx]; aR += REP[idx]; aM += AM[idx]; aN += RM[idx];
  }
  sA[t] = aA; sR[t] = aR; sM[t] = aM; sN[t] = aN;
  __syncthreads();
  for (int s = 512; s > 0; s >>= 1) {             // fixed-order tree
    if (t < s) { sA[t] += sA[t+s]; sR[t] += sR[t+s]; sM[t] += sM[t+s]; sN[t] += sN[t+s]; }
    __syncthreads();
  }
  if (t == 0) {
    float att_loss = sA[0] / (sM[0] + 1e-10f);
    att_loss = (att_loss < 0.0f) ? 0.0f : att_loss;   // ATT_LOSS_GATE = 0
    float rep_loss = sR[0] / (sN[0] + 1e-10f);
    rep_loss = (rep_loss < 0.0f) ? 0.0f : rep_loss;   // REP_LOSS_GATE = 0
    out[0] = att_loss;
    out[1] = rep_loss;
  }
}

extern "C" void kernel_launch(void* const* d_in, const int* in_sizes, int n_in,
                              void* d_out, int out_size, void* d_ws, size_t ws_size,
                              hipStream_t stream) {
  const float* prior = (const float*)d_in[0];   // (8,64,32,32)
  const float* h     = (const float*)d_in[1];   // (8,64,1024)
  float* out = (float*)d_out;                   // 2 scalars

  // workspace layout (floats)
  float* ws  = (float*)d_ws;
  float* P   = ws;                               // 524288
  float* LP  = P  + B_ * L_ * K_ * C_;           // 524288
  float* SQ  = LP + B_ * L_ * K_ * C_;           // 512
  float* G   = SQ + B_ * L_;                     // 32768
  float* ATT = G   + PAIRS;                      // 32768
  float* REP = ATT + PAIRS;
  float* AM  = REP + PAIRS;
  float* RM  = AM  + PAIRS;

  // K1: softmax rows = 16384, 8 waves/block -> 2048 blocks
  k_softmax<<<2048, 256, 0, stream>>>(prior, P, LP);
  // K2: 512 rows, 8 waves/block -> 64 blocks
  k_sqnorm<<<64, 256, 0, stream>>>(h, SQ);
  // K3: 128 single-wave tiles (8 batches x 4x4)
  k_gram_wmma<<<128, 32, 0, stream>>>(h, G);
  // K4: 32768 pairs, 8 waves/block -> 4096 blocks
  k_pairs<<<4096, 256, 0, stream>>>(P, LP, SQ, G, ATT, REP, AM, RM);
  // K5: single block deterministic reduction
  k_reduce<<<1, 1024, 0, stream>>>(ATT, REP, AM, RM, out);
}